// AdaptSelfAttention_36395552866534
// MI455X (gfx1250) — compile-verified
//
#include <hip/hip_runtime.h>

#define B_   4
#define S_   320
#define H_   512
#define NH_  8
#define DH_  64
#define NEGV (-1e15f)

typedef float v2f __attribute__((ext_vector_type(2)));
typedef float v8f __attribute__((ext_vector_type(8)));

// D = A(16x4) * B(4x16) + C, all f32, wave32.
__device__ __forceinline__ v8f wmma4(v2f a, v2f b, v8f c) {
  return __builtin_amdgcn_wmma_f32_16x16x4_f32(
      /*neg_a=*/false, a, /*neg_b=*/false, b,
      /*c_mod=*/(short)0, c, /*reuse_a=*/false, /*reuse_b=*/false);
}

// ---------------------------------------------------------------------------
// K1: qproj = query @ Wq^T + bq ; vproj = value @ Wv^T + bv
// grid (N/16=32, M/16=80, 2), block 32 (one wave per 16x16 tile, K=512)
// ---------------------------------------------------------------------------
__global__ __launch_bounds__(32) void proj_kernel(
    const float* __restrict__ query, const float* __restrict__ value,
    const float* __restrict__ Wq, const float* __restrict__ bq,
    const float* __restrict__ Wv, const float* __restrict__ bv,
    float* __restrict__ qproj, float* __restrict__ vproj) {
  const int lane = threadIdx.x;
  const int n0 = blockIdx.x * 16;
  const int m0 = blockIdx.y * 16;
  const bool doV = (blockIdx.z != 0);
  const float* A    = doV ? value : query;
  const float* W    = doV ? Wv : Wq;
  const float* bias = doV ? bv : bq;
  float* out        = doV ? vproj : qproj;

  const int r  = lane & 15;   // A: M row / B: N col
  const int hi = lane >> 4;   // K half selector
  v8f acc = {};
  const float* arow = A + (size_t)(m0 + r) * H_ + 2 * hi;
  // out[m,n] = sum_k A[m,k] * W[n,k]  ->  B[kk,n] = W[n0+n, k0+kk]
  const float* brow = W + (size_t)(n0 + r) * H_ + 2 * hi;
  for (int k = 0; k < H_; k += 4) {
    v2f a  = *(const v2f*)(arow + k);
    v2f bb = *(const v2f*)(brow + k);
    acc = wmma4(a, bb, acc);
  }
  const float bval = bias[n0 + r];
  for (int j = 0; j < 8; ++j) {
    const int m = m0 + 8 * hi + j;
    out[(size_t)m * H_ + n0 + r] = acc[j] + bval;
  }
}

// ---------------------------------------------------------------------------
// K2: P[m, h, e] = sum_d (qproj[m, h*64+d] + v[h*64+d]) * Wr[h*64+d, e]
// grid (E/16=32, M/16=80, NH=8), block 32, K=64
// ---------------------------------------------------------------------------
__global__ __launch_bounds__(32) void pproj_kernel(
    const float* __restrict__ qproj, const float* __restrict__ Wr,
    const float* __restrict__ vvec, float* __restrict__ P) {
  const int lane = threadIdx.x;
  const int e0 = blockIdx.x * 16;
  const int m0 = blockIdx.y * 16;
  const int h  = blockIdx.z;
  const int r  = lane & 15;
  const int hi = lane >> 4;
  v8f acc = {};
  const float* arow = qproj + (size_t)(m0 + r) * H_ + h * DH_ + 2 * hi;
  const float* vv   = vvec + h * DH_ + 2 * hi;
  for (int k = 0; k < DH_; k += 4) {
    v2f a  = *(const v2f*)(arow + k);
    v2f av = *(const v2f*)(vv + k);
    a = a + av;
    // B[kk, n] = Wr[h*64 + k + kk, e0 + n]
    const float b0 = Wr[(size_t)(h * DH_ + k + 2 * hi + 0) * H_ + e0 + r];
    const float b1 = Wr[(size_t)(h * DH_ + k + 2 * hi + 1) * H_ + e0 + r];
    v2f bb; bb.x = b0; bb.y = b1;
    acc = wmma4(a, bb, acc);
  }
  for (int j = 0; j < 8; ++j) {
    const int m = m0 + 8 * hi + j;
    P[(size_t)m * (NH_ * H_) + h * H_ + e0 + r] = acc[j];
  }
}

// ---------------------------------------------------------------------------
// K3: fused scores + mask + softmax + attn@vproj. One block per (b,q).
// scores[h,k] = sum_e rpe[b,q,k,e]*P[b,h,q,e]
//             + sum_d key[b,k,h*64+d]*qu[b,h,q,d] + c[b,h,q] + rand[h,q,k]
// block = 320 threads (10 waves); each wave does 2 key tiles of 16.
// ---------------------------------------------------------------------------
__global__ __launch_bounds__(320) void attn_kernel(
    const float* __restrict__ key,  const float* __restrict__ rpe,
    const float* __restrict__ qproj,const float* __restrict__ vproj,
    const float* __restrict__ P,    const float* __restrict__ uvec,
    const float* __restrict__ vvec, const float* __restrict__ brv,
    const float* __restrict__ randA,const int* __restrict__ seq_len,
    const int* __restrict__ lexn,   float* __restrict__ out) {
  __shared__ float ldsB[NH_ * 2 * H_];   // B-matrix: [h][e=0..1023], 32 KB
  __shared__ float ldsS[NH_ * S_];       // scores/exp: [h][k], 10 KB
  __shared__ float ldsC[NH_];
  __shared__ float ldsDen[NH_];

  const int bq  = blockIdx.x;
  const int b   = bq / S_;
  const int q   = bq % S_;
  const int tid = threadIdx.x;
  const int lane = tid & 31;
  const int w    = tid >> 5;
  const int len  = seq_len[b] + lexn[0];

  const float* Prow = P + (size_t)bq * (NH_ * H_);
  const float* qrow = qproj + (size_t)bq * H_;

  // ---- phase A: build B-matrix in LDS ----
  for (int i = tid; i < NH_ * H_; i += 320) {
    const int h = i >> 9;
    const int e = i & (H_ - 1);
    ldsB[h * 1024 + e]        = Prow[i];   // Prow is [h][e]
    ldsB[h * 1024 + 512 + e]  = 0.0f;
  }
  __syncthreads();
  for (int i = tid; i < H_; i += 320) {    // block-diagonal qu part
    const int h = i >> 6;                  // i = h*64 + d
    ldsB[h * 1024 + 512 + i] = qrow[i] + uvec[i];
  }
  if (tid < NH_) {
    float c = 0.f;
    for (int d = 0; d < DH_; ++d)
      c += (qrow[tid * DH_ + d] + vvec[tid * DH_ + d]) * brv[tid * DH_ + d];
    ldsC[tid] = c;
  }
  __syncthreads();

  // ---- phase B: scores via WMMA ----
  const int r  = lane & 15;
  const int hi = lane >> 4;
  const float* rbase = rpe + (size_t)bq * S_ * H_;
  const float* kbase = key + (size_t)b * S_ * H_;
  for (int t = 0; t < 2; ++t) {
    const int kt  = w * 2 + t;       // key tile 0..19
    const int row = kt * 16 + r;
    v8f acc = {};
    const int nn = (r < 8) ? r : 0;  // clamp LDS address; zero data for n>=8
    // part 1: rpe rows, e in [0,512)
    {
      const float* ar = rbase + (size_t)row * H_ + 2 * hi;
      const float* bp = &ldsB[nn * 1024 + 2 * hi];
      for (int kk = 0; kk < H_; kk += 4) {
        v2f a  = *(const v2f*)(ar + kk);
        v2f bb = *(const v2f*)(bp + kk);
        if (r >= 8) { bb.x = 0.f; bb.y = 0.f; }
        acc = wmma4(a, bb, acc);
      }
    }
    // part 2: key rows, e in [512,1024) against block-diag qu
    {
      const float* ak = kbase + (size_t)row * H_ + 2 * hi;
      const float* bp = &ldsB[nn * 1024 + 512 + 2 * hi];
      for (int kk = 0; kk < H_; kk += 4) {
        v2f a  = *(const v2f*)(ak + kk);
        v2f bb = *(const v2f*)(bp + kk);
        if (r >= 8) { bb.x = 0.f; bb.y = 0.f; }
        acc = wmma4(a, bb, acc);
      }
    }
    if (r < 8) {   // D layout: col N=r (head), rows M = kt*16 + 8*hi + j
      const float cc = ldsC[r];
      for (int j = 0; j < 8; ++j) {
        const int k = kt * 16 + 8 * hi + j;
        float sv = acc[j] + cc + randA[((size_t)r * S_ + q) * S_ + k];
        if (k >= len) sv = NEGV;
        ldsS[r * S_ + k] = sv;
      }
    }
  }
  __syncthreads();

  // ---- phase C: softmax, one wave per head ----
  if (w < NH_) {
    const int h = w;
    float m = -3.4e38f;
    for (int k = lane; k < S_; k += 32) m = fmaxf(m, ldsS[h * S_ + k]);
    for (int off = 16; off > 0; off >>= 1) m = fmaxf(m, __shfl_xor(m, off, 32));
    float s = 0.f;
    for (int k = lane; k < S_; k += 32) {
      const float e = __expf(ldsS[h * S_ + k] - m);
      ldsS[h * S_ + k] = e;
      s += e;
    }
    for (int off = 16; off > 0; off >>= 1) s += __shfl_xor(s, off, 32);
    if (lane == 0) ldsDen[h] = s;
  }
  __syncthreads();

  // ---- phase D: out[b,q,h*64+d] = (attn @ vproj) / denom ----
  const float* vb = vproj + (size_t)b * S_ * H_;
  for (int i = tid; i < H_; i += 320) {
    const int h = i >> 6;
    float s = 0.f;
    for (int k = 0; k < S_; ++k)
      s += ldsS[h * S_ + k] * vb[(size_t)k * H_ + i];
    out[(size_t)bq * H_ + i] = s / ldsDen[h];
  }
}

// ---------------------------------------------------------------------------
extern "C" void kernel_launch(void* const* d_in, const int* in_sizes, int n_in,
                              void* d_out, int out_size, void* d_ws, size_t ws_size,
                              hipStream_t stream) {
  const float* query = (const float*)d_in[0];
  const float* key   = (const float*)d_in[1];
  const float* value = (const float*)d_in[2];
  const float* rpe   = (const float*)d_in[3];
  const int*   slen  = (const int*)d_in[4];
  const int*   lexn  = (const int*)d_in[5];
  const float* Wq    = (const float*)d_in[6];
  const float* bq    = (const float*)d_in[7];
  const float* Wv    = (const float*)d_in[8];
  const float* bv    = (const float*)d_in[9];
  const float* Wr    = (const float*)d_in[10];
  const float* br    = (const float*)d_in[11];
  const float* uvec  = (const float*)d_in[12];
  const float* vvec  = (const float*)d_in[13];
  const float* randA = (const float*)d_in[14];
  float* out = (float*)d_out;

  float* qproj = (float*)d_ws;                       // [1280,512]
  float* vproj = qproj + (size_t)B_ * S_ * H_;       // [1280,512]
  float* P     = vproj + (size_t)B_ * S_ * H_;       // [1280, 8, 512]

  proj_kernel<<<dim3(H_ / 16, (B_ * S_) / 16, 2), 32, 0, stream>>>(
      query, value, Wq, bq, Wv, bv, qproj, vproj);
  pproj_kernel<<<dim3(H_ / 16, (B_ * S_) / 16, NH_), 32, 0, stream>>>(
      qproj, Wr, vvec, P);
  attn_kernel<<<dim3(B_ * S_), 320, 0, stream>>>(
      key, rpe, qproj, vproj, P, uvec, vvec, br, randA, slen, lexn, out);
}